// GCNDecoder_10101763080452
// MI455X (gfx1250) — compile-verified
//
#include <hip/hip_runtime.h>
#include <hip/hip_bf16.h>
#include <math.h>

#define N_NODES 100000
#define N_EDGES 1600000
#define EPS_BN 1e-5f
#define ALPHA_LRELU 0.01f

typedef __attribute__((ext_vector_type(2))) float v2f;
typedef __attribute__((ext_vector_type(8))) float v8f;

// ---------------------------------------------------------------------------
// Generic fill
// ---------------------------------------------------------------------------
__global__ void gcn_fill(float* __restrict__ p, float v, int n) {
    int i = blockIdx.x * blockDim.x + threadIdx.x;
    if (i < n) p[i] = v;
}

// deg[dst[e]] += 1  (deg pre-filled with 1.0 for the self loop)
__global__ void gcn_deg_accum(const int* __restrict__ dst, float* __restrict__ deg, int E) {
    int e = blockIdx.x * blockDim.x + threadIdx.x;
    if (e < E) atomicAdd(&deg[dst[e]], 1.0f);
}

// deg -> rsqrt(deg) in place
__global__ void gcn_rsqrt_inplace(float* __restrict__ d, int n) {
    int i = blockIdx.x * blockDim.x + threadIdx.x;
    if (i < n) d[i] = rsqrtf(d[i]);
}

// ---------------------------------------------------------------------------
// WMMA f32 GEMM, compile-time N and K:
//   C[M,N] = A[M,K] @ B[K,N] (+ bias[N] if bias != nullptr)
// One wave computes a 16x64 C stripe: 4 accumulators share each A fragment,
// K stepped by 4 via V_WMMA_F32_16X16X4_F32. Constant strides -> immediate
// vmem offsets instead of per-load 64-bit address math.
// ---------------------------------------------------------------------------
template <int N, int K>
__global__ void gcn_wmma_gemm(const float* __restrict__ A,
                              const float* __restrict__ B,
                              const float* __restrict__ bias,
                              float* __restrict__ C, int M) {
    constexpr int TILES_N = N / 64;           // column groups of 64 per wave
    const int lane = threadIdx.x & 31;
    const int wave = (blockIdx.x * blockDim.x + threadIdx.x) >> 5;
    const int tm   = wave / TILES_N;
    const int tn   = wave - tm * TILES_N;
    if (tm >= M / 16) return;                 // whole-wave exit: EXEC all-1s

    const int half = lane >> 4;               // 0: lanes 0-15, 1: lanes 16-31
    const int l15  = lane & 15;
    const int m    = tm * 16 + l15;           // A row for this lane
    const int kb   = half * 2;                // K sub-pair per lane half
    const int n0   = tn * 64 + l15;           // first of 4 columns (j*16 apart)

    const float* __restrict__ Arow = A + (size_t)m * K + kb;
    const float* __restrict__ Bp   = B + (size_t)kb * N + n0;

    v8f acc[4] = {};
    #pragma unroll 4
    for (int k = 0; k < K; k += 4) {
        v2f a;
        a.x = Arow[k];
        a.y = Arow[k + 1];
        const float* __restrict__ bk = Bp + (size_t)k * N;
        #pragma unroll
        for (int j = 0; j < 4; ++j) {
            v2f b;
            b.x = bk[j * 16];                 // row k+kb,   col n0+j*16
            b.y = bk[N + j * 16];             // row k+kb+1, col n0+j*16
            acc[j] = __builtin_amdgcn_wmma_f32_16x16x4_f32(
                         false, a, false, b, (short)0, acc[j], false, false);
        }
    }

    const int mbase = tm * 16 + half * 8;     // C/D: VGPR v holds row mbase+v
    #pragma unroll
    for (int j = 0; j < 4; ++j) {
        const int n = n0 + j * 16;
        float bv = bias ? bias[n] : 0.0f;
        #pragma unroll
        for (int v = 0; v < 8; ++v)
            C[(size_t)(mbase + v) * N + n] = acc[j][v] + bv;
    }
}

// ---------------------------------------------------------------------------
// agg[n,:] = H[n,:] * dinv[n]^2   (self-loop seed), float4-vectorized
// ---------------------------------------------------------------------------
__global__ void gcn_self_loop_init(const float4* __restrict__ H,
                                   const float* __restrict__ dinv,
                                   float4* __restrict__ agg,
                                   int n_nodes, int Fv) {       // Fv = F/4
    size_t i = (size_t)blockIdx.x * blockDim.x + threadIdx.x;
    size_t total = (size_t)n_nodes * Fv;
    if (i >= total) return;
    int node = (int)(i / Fv);
    float di = dinv[node];
    float s = di * di;
    float4 h = H[i];
    agg[i] = make_float4(h.x * s, h.y * s, h.z * s, h.w * s);
}

// ---------------------------------------------------------------------------
// One wave per edge: agg[dst,:] += H[src,:] * dinv[src]*dinv[dst]
// Lanes stride the feature dim -> coalesced loads and coalesced f32 atomics.
// ---------------------------------------------------------------------------
__global__ void gcn_edge_scatter(const int* __restrict__ src,
                                 const int* __restrict__ dst,
                                 const float* __restrict__ dinv,
                                 const float* __restrict__ H,
                                 float* __restrict__ agg,
                                 int E, int F) {
    int wave = (blockIdx.x * blockDim.x + threadIdx.x) >> 5;
    int lane = threadIdx.x & 31;
    if (wave >= E) return;
    int s = src[wave];
    int d = dst[wave];
    float nrm = dinv[s] * dinv[d];
    const float* __restrict__ hs = H + (size_t)s * F;
    float* __restrict__ ad = agg + (size_t)d * F;
    #pragma unroll 2
    for (int f = lane; f < F; f += 32)
        atomicAdd(&ad[f], hs[f] * nrm);
}

// ---------------------------------------------------------------------------
// BatchNorm column stats: blockDim.x == F, thread t owns column t.
// Each block covers a row stripe; partials merged via global f32 atomics.
// ---------------------------------------------------------------------------
__global__ void gcn_bn_stats(const float* __restrict__ x,
                             float* __restrict__ sum,
                             float* __restrict__ sumsq,
                             int n_rows, int F) {
    int f = threadIdx.x;
    float s = 0.0f, ss = 0.0f;
    for (int r = blockIdx.x; r < n_rows; r += gridDim.x) {
        float v = x[(size_t)r * F + f];
        s  += v;
        ss += v * v;
    }
    atomicAdd(&sum[f], s);
    atomicAdd(&sumsq[f], ss);
}

__global__ void gcn_bn_finalize(const float* __restrict__ sum,
                                const float* __restrict__ sumsq,
                                float* __restrict__ mu,
                                float* __restrict__ rstd,
                                int F, float invN) {
    int f = blockIdx.x * blockDim.x + threadIdx.x;
    if (f >= F) return;
    float m   = sum[f] * invN;
    float var = sumsq[f] * invN - m * m;
    mu[f]   = m;
    rstd[f] = rsqrtf(var + EPS_BN);
}

// x = leaky_relu((x - mu) * rstd * gamma + beta)   in place
__global__ void gcn_bn_apply_leaky(float* __restrict__ x,
                                   const float* __restrict__ mu,
                                   const float* __restrict__ rstd,
                                   const float* __restrict__ g,
                                   const float* __restrict__ beta,
                                   size_t total, int Fmask) {
    size_t i = (size_t)blockIdx.x * blockDim.x + threadIdx.x;
    if (i >= total) return;
    int f = (int)(i & (size_t)Fmask);         // F is a power of two (128/256)
    float v = (x[i] - mu[f]) * rstd[f] * g[f] + beta[f];
    x[i] = v >= 0.0f ? v : ALPHA_LRELU * v;
}

// ---------------------------------------------------------------------------
// Host-side launch sequence
// ---------------------------------------------------------------------------
static inline int cdiv(long long a, long long b) { return (int)((a + b - 1) / b); }

extern "C" void kernel_launch(void* const* d_in, const int* in_sizes, int n_in,
                              void* d_out, int out_size, void* d_ws, size_t ws_size,
                              hipStream_t stream) {
    (void)in_sizes; (void)n_in; (void)out_size; (void)ws_size;

    const float* emb   = (const float*)d_in[0];
    const int*   eidx  = (const int*)  d_in[1];
    const float* W1    = (const float*)d_in[2];
    // b1 (d_in[3]) cancels under BatchNorm — mathematically exact to skip.
    const float* g1    = (const float*)d_in[4];
    const float* beta1 = (const float*)d_in[5];
    const float* W2    = (const float*)d_in[6];
    // b2 (d_in[7]) cancels under BatchNorm.
    const float* g2    = (const float*)d_in[8];
    const float* beta2 = (const float*)d_in[9];
    const float* Wf    = (const float*)d_in[10];
    const float* bf    = (const float*)d_in[11];
    float* out = (float*)d_out;

    const int* src = eidx;              // edge_index[0]
    const int* dst = eidx + N_EDGES;    // edge_index[1]

    // Workspace layout (floats)
    float* ws    = (float*)d_ws;
    float* dinv  = ws;                       // N_NODES
    float* sum   = ws + 100352;              // 256
    float* sumsq = sum + 256;                // 256
    float* mu    = sumsq + 256;              // 256
    float* rstd  = mu + 256;                 // 256
    float* H     = ws + 102400;              // 100000*256
    float* AGG   = H + (size_t)N_NODES * 256;// 100000*256

    const float invN = 1.0f / (float)N_NODES;

    // ---- degrees over A + I, dinv = rsqrt(deg) ----
    gcn_fill<<<cdiv(N_NODES, 256), 256, 0, stream>>>(dinv, 1.0f, N_NODES);
    gcn_deg_accum<<<cdiv(N_EDGES, 256), 256, 0, stream>>>(dst, dinv, N_EDGES);
    gcn_rsqrt_inplace<<<cdiv(N_NODES, 256), 256, 0, stream>>>(dinv, N_NODES);

    // ================= Layer 1: F = 128, K = 64 =================
    {
        const int M = N_NODES, N = 128;
        long long waves = (long long)(M / 16) * (N / 64);
        gcn_wmma_gemm<128, 64><<<cdiv(waves * 32, 256), 256, 0, stream>>>(
            emb, W1, nullptr, H, M);
        size_t total = (size_t)M * N;
        gcn_self_loop_init<<<cdiv(total / 4, 256), 256, 0, stream>>>(
            (const float4*)H, dinv, (float4*)AGG, M, N / 4);
        gcn_edge_scatter<<<cdiv((long long)N_EDGES * 32, 256), 256, 0, stream>>>(
            src, dst, dinv, H, AGG, N_EDGES, N);
        gcn_fill<<<2, 256, 0, stream>>>(sum, 0.0f, 512);   // sum + sumsq contiguous
        gcn_bn_stats<<<1024, N, 0, stream>>>(AGG, sum, sumsq, M, N);
        gcn_bn_finalize<<<1, 256, 0, stream>>>(sum, sumsq, mu, rstd, N, invN);
        gcn_bn_apply_leaky<<<cdiv(total, 256), 256, 0, stream>>>(
            AGG, mu, rstd, g1, beta1, total, N - 1);
    }

    // ================= Layer 2: F = 256, K = 128 =================
    {
        const int M = N_NODES, N = 256;
        long long waves = (long long)(M / 16) * (N / 64);
        gcn_wmma_gemm<256, 128><<<cdiv(waves * 32, 256), 256, 0, stream>>>(
            AGG, W2, nullptr, H, M);
        size_t total = (size_t)M * N;
        gcn_self_loop_init<<<cdiv(total / 4, 256), 256, 0, stream>>>(
            (const float4*)H, dinv, (float4*)AGG, M, N / 4);
        gcn_edge_scatter<<<cdiv((long long)N_EDGES * 32, 256), 256, 0, stream>>>(
            src, dst, dinv, H, AGG, N_EDGES, N);
        gcn_fill<<<2, 256, 0, stream>>>(sum, 0.0f, 512);
        gcn_bn_stats<<<1024, N, 0, stream>>>(AGG, sum, sumsq, M, N);
        gcn_bn_finalize<<<1, 256, 0, stream>>>(sum, sumsq, mu, rstd, N, invN);
        gcn_bn_apply_leaky<<<cdiv(total, 256), 256, 0, stream>>>(
            AGG, mu, rstd, g2, beta2, total, N - 1);
    }

    // ================= Final linear: out = act2 @ Wf + bf =================
    {
        const int M = N_NODES, N = 256;
        long long waves = (long long)(M / 16) * (N / 64);
        gcn_wmma_gemm<256, 256><<<cdiv(waves * 32, 256), 256, 0, stream>>>(
            AGG, Wf, bf, out, M);
    }
}